// GAE_51539607552044
// MI455X (gfx1250) — compile-verified
//
#include <hip/hip_runtime.h>
#include <math.h>

// ---------------------------------------------------------------------------
// GCN autoencoder anomaly detection for MI455X (gfx1250, wave32).
// Memory-bound (≈3.5 GB irregular traffic @ 23.3 TB/s); GEMMs are ~5 GFLOP,
// done with full-precision V_WMMA_F32_16X16X4_F32. W1/W2 are pre-transposed
// so B fragments are contiguous float2 loads with immediate offsets, and the
// GEMM is templated on NCOLS so all strides are compile-time constants.
// ---------------------------------------------------------------------------

typedef float v2f __attribute__((ext_vector_type(2)));
typedef float v8f __attribute__((ext_vector_type(8)));

__device__ __forceinline__ void atomAddF(float* p, float v) {
    __hip_atomic_fetch_add(p, v, __ATOMIC_RELAXED, __HIP_MEMORY_SCOPE_AGENT);
}

// ---- setup: deg starts at 1.0 (self loop), err/cnt zeroed ------------------
__global__ void k_init(float* __restrict__ deg, float* __restrict__ errsum,
                       float* __restrict__ cnt, int N) {
    int i = blockIdx.x * blockDim.x + threadIdx.x;
    if (i < N) { deg[i] = 1.0f; errsum[i] = 0.0f; cnt[i] = 0.0f; }
}

__global__ void k_degree(const int* __restrict__ dst, float* __restrict__ deg, int E) {
    int e = blockIdx.x * blockDim.x + threadIdx.x;
    if (e < E) atomAddF(&deg[dst[e]], 1.0f);
}

__global__ void k_rsqrt(float* __restrict__ deg, int N) {
    int i = blockIdx.x * blockDim.x + threadIdx.x;
    if (i < N) deg[i] = rsqrtf(deg[i]);
}

// Wt[n*K + k] = W[k*N + n]   (K = 128 rows, N columns)
__global__ void k_transpose(const float* __restrict__ W, float* __restrict__ Wt,
                            int N) {
    int idx = blockIdx.x * blockDim.x + threadIdx.x;   // over 128*N
    if (idx < 128 * N) {
        int k = idx / N, n = idx - k * N;
        Wt[n * 128 + k] = W[idx];
    }
}

// out[i*dim + j] = b[j]   (dim is a power of two: 128 or 64)
__global__ void k_fill_bias(float* __restrict__ out, const float* __restrict__ b,
                            int total, int dimMask) {
    int i = blockIdx.x * blockDim.x + threadIdx.x;
    if (i < total) out[i] = b[i & dimMask];
}

__global__ void k_relu(float* __restrict__ a, int total) {
    int i = blockIdx.x * blockDim.x + threadIdx.x;
    if (i < total) a[i] = fmaxf(a[i], 0.0f);
}

// ---- fp32 WMMA GEMM: C[M x NCOLS] = A[M x 128] * Bt^T ---------------------
// Bt is NCOLS x 128 (pre-transposed weights: row n holds column n of W).
// Grid: M/16 blocks (M % 16 == 0; M = 100000 = 16 * 6250).
// blockDim.x = (NCOLS/16)*32, each wave owns one 16-col slice; EXEC all-1s.
// A tile (16x128) staged in LDS with 132-float row stride (4-bank skew).
template <int NCOLS>
__global__ void k_gemm_wmma_k128(const float* __restrict__ A,
                                 const float* __restrict__ Bt,
                                 float* __restrict__ C) {
    __shared__ float As[16 * 132];
    const int tid  = threadIdx.x;
    const int row0 = blockIdx.x * 16;

    // cooperative stage of the 16x128 A tile (512 float4 total)
    const float4* Ag = (const float4*)(A + (size_t)row0 * 128);
    for (int i = tid; i < 16 * 32; i += blockDim.x) {
        int r = i >> 5, c4 = i & 31;
        ((float4*)(As + r * 132))[c4] = Ag[r * 32 + c4];
    }
    __syncthreads();

    const int lane  = tid & 31;
    const int col0  = (tid >> 5) * 16;
    const int n15   = lane & 15;        // N index (and A row index for frag A)
    const int khalf = lane >> 4;        // selects K pair within a step

    v8f acc = {0.f, 0.f, 0.f, 0.f, 0.f, 0.f, 0.f, 0.f};
    // base pointers hoisted; unrolled kk becomes an immediate offset
    const float* ap = As + n15 * 132 + 2 * khalf;
    const float* bp = Bt + (size_t)(col0 + n15) * 128 + 2 * khalf;

#pragma unroll
    for (int kk = 0; kk < 128; kk += 4) {
        // A 16x4 frag: lane<16 -> K=kk,kk+1 ; lane>=16 -> K=kk+2,kk+3
        v2f a; a.x = ap[kk]; a.y = ap[kk + 1];
        // B 4x16 frag: lane = column, khalf selects K half (contiguous in Bt)
        v2f b = *(const v2f*)(bp + kk);
        acc = __builtin_amdgcn_wmma_f32_16x16x4_f32(
            /*neg_a=*/false, a, /*neg_b=*/false, b,
            /*c_mod=*/(short)0, acc, /*reuse_a=*/false, /*reuse_b=*/false);
    }

    // C/D layout: VGPR r, lane<16 -> row r ; lane>=16 -> row 8+r
    float* Cp = C + (size_t)(row0 + 8 * khalf) * NCOLS + col0 + n15;
#pragma unroll
    for (int r = 0; r < 8; ++r) Cp[r * NCOLS] = acc[r];
}

// ---- message aggregation: one wave per (edge or self-loop) ----------------
// out[dst] += dinv[src]*dinv[dst] * H[src]   over E edges + N self loops.
template <int DIM>
__global__ void k_aggregate(const float* __restrict__ H, float* __restrict__ out,
                            const int* __restrict__ src, const int* __restrict__ dst,
                            const float* __restrict__ dinv, int E, int N) {
    const int  lane  = threadIdx.x & 31;
    const long long w = (long long)blockIdx.x * (blockDim.x >> 5) + (threadIdx.x >> 5);
    const long long total = (long long)E + N;
    if (w >= total) return;

    int s, d;
    if (w < E) { s = src[w]; d = dst[w]; }
    else       { s = d = (int)(w - E); }
    const float norm = dinv[s] * dinv[d];

    constexpr int VPL = DIM / 32;                 // floats per lane: 4 or 2
    const float* hp = H   + (size_t)s * DIM + lane * VPL;
    float*       op = out + (size_t)d * DIM + lane * VPL;
    if constexpr (VPL == 4) {
        float4 v = *(const float4*)hp;
        atomAddF(op + 0, v.x * norm);
        atomAddF(op + 1, v.y * norm);
        atomAddF(op + 2, v.z * norm);
        atomAddF(op + 3, v.w * norm);
    } else {
        float2 v = *(const float2*)hp;
        atomAddF(op + 0, v.x * norm);
        atomAddF(op + 1, v.y * norm);
    }
}

// ---- edge decode: wave per edge, 64-dim dot + wave32 reduction ------------
__global__ void k_decode(const float* __restrict__ Z, const int* __restrict__ row,
                         const int* __restrict__ col, float* __restrict__ errsum,
                         float* __restrict__ cnt, int E) {
    const int lane = threadIdx.x & 31;
    const long long e = (long long)blockIdx.x * (blockDim.x >> 5) + (threadIdx.x >> 5);
    if (e >= E) return;
    const int r = row[e], c = col[e];

    const float2 zr = *(const float2*)(Z + (size_t)r * 64 + lane * 2);
    const float2 zc = *(const float2*)(Z + (size_t)c * 64 + lane * 2);
    float p = zr.x * zc.x + zr.y * zc.y;
#pragma unroll
    for (int off = 16; off > 0; off >>= 1) p += __shfl_xor(p, off, 32);

    if (lane == 0) {
        float sig = 1.0f / (1.0f + __expf(-p));
        float err = -__logf(sig + 1e-15f);
        atomAddF(&errsum[r], err);
        atomAddF(&cnt[r], 1.0f);
    }
}

__global__ void k_finalize(const float* __restrict__ errsum, const float* __restrict__ cnt,
                           float* __restrict__ out, int N) {
    int i = blockIdx.x * blockDim.x + threadIdx.x;
    if (i < N) {
        float c = cnt[i];
        out[i] = (c > 0.0f) ? (errsum[i] / fmaxf(c, 1.0f)) : 10.0f;
    }
}

// ---------------------------------------------------------------------------
extern "C" void kernel_launch(void* const* d_in, const int* in_sizes, int n_in,
                              void* d_out, int out_size, void* d_ws, size_t ws_size,
                              hipStream_t stream) {
    const float* x  = (const float*)d_in[0];
    const int*   ei = (const int*)d_in[1];   // JAX x64 disabled -> int32 indices
    const float* W1 = (const float*)d_in[2];
    const float* b1 = (const float*)d_in[3];
    const float* W2 = (const float*)d_in[4];
    const float* b2 = (const float*)d_in[5];

    const int N = in_sizes[0] / 128;   // 100000 (divisible by 16)
    const int E = in_sizes[1] / 2;     // 1600000
    const int* src = ei;
    const int* dst = ei + E;

    // workspace layout (floats), 256B-aligned slabs; bufA/bufB reused
    float* ws     = (float*)d_ws;
    float* dinv   = ws;                                   // N  (deg -> rsqrt in place)
    float* errsum = dinv + N;                             // N
    float* cnt    = errsum + N;                           // N
    size_t off    = (((size_t)3 * N) + 63) & ~(size_t)63;
    float* wt1    = ws + off;                             // 128*128 (W1^T)
    float* wt2    = wt1 + 128 * 128;                      // 64*128  (W2^T)
    float* bufA   = wt2 + 64 * 128;                       // N*128 : h0, then z0
    float* bufB   = bufA + (size_t)N * 128;               // N*128 : h(agg1), then z(agg2)

    const int T = 256;
    // degree / normalization + weight transposes (tiny, one-shot)
    k_init     <<<(N + T - 1) / T, T, 0, stream>>>(dinv, errsum, cnt, N);
    k_degree   <<<(E + T - 1) / T, T, 0, stream>>>(dst, dinv, E);
    k_rsqrt    <<<(N + T - 1) / T, T, 0, stream>>>(dinv, N);
    k_transpose<<<(128 * 128 + T - 1) / T, T, 0, stream>>>(W1, wt1, 128);
    k_transpose<<<(128 * 64  + T - 1) / T, T, 0, stream>>>(W2, wt2, 64);

    // layer 1: h0 = x @ W1 ; agg1 = b1 + scatter(norm * h0) ; relu
    k_gemm_wmma_k128<128><<<N / 16, 256, 0, stream>>>(x, wt1, bufA);
    k_fill_bias<<<(N * 128 + T - 1) / T, T, 0, stream>>>(bufB, b1, N * 128, 127);
    k_aggregate<128><<<(E + N + 7) / 8, 256, 0, stream>>>(bufA, bufB, src, dst, dinv, E, N);
    k_relu<<<(N * 128 + T - 1) / T, T, 0, stream>>>(bufB, N * 128);

    // layer 2: z0 = h @ W2 ; z = b2 + scatter(norm * z0)
    k_gemm_wmma_k128<64><<<N / 16, 128, 0, stream>>>(bufB, wt2, bufA);
    k_fill_bias<<<(N * 64 + T - 1) / T, T, 0, stream>>>(bufB, b2, N * 64, 63);
    k_aggregate<64><<<(E + N + 7) / 8, 256, 0, stream>>>(bufA, bufB, src, dst, dinv, E, N);

    // decode + per-source anomaly
    k_decode<<<(E + 7) / 8, 256, 0, stream>>>(bufB, src, dst, errsum, cnt, E);
    k_finalize<<<(N + T - 1) / T, T, 0, stream>>>(errsum, cnt, (float*)d_out, N);
}